// ScaledDotAttention_54073638256901
// MI455X (gfx1250) — compile-verified
//
#include <hip/hip_runtime.h>

#define BATCH   64
#define SEQLEN  8192
#define DQ      128
#define DV      128
#define CHUNK   1024
#define NCHUNK  (SEQLEN / CHUNK)          // 8
#define WAVES   8                          // 256 threads / 32
#define TILES_PER_WAVE (CHUNK / (WAVES * 16))  // 8

typedef __attribute__((ext_vector_type(2))) float v2f;
typedef __attribute__((ext_vector_type(8))) float v8f;

// ---------------------------------------------------------------------------
// Kernel 1: raw scaled scores via V_WMMA_F32_16X16X4_F32, per-chunk max.
// grid = (BATCH, NCHUNK), block = 256 (8 waves). Each wave: 8 tiles of 16 keys.
// ---------------------------------------------------------------------------
__global__ __launch_bounds__(256) void
sda_scores_kernel(const float* __restrict__ q, const float* __restrict__ k,
                  float* __restrict__ scores_out, float* __restrict__ chunk_max)
{
    __shared__ float qs[DQ];
    __shared__ float wmax[WAVES];

    const int b    = blockIdx.x;
    const int c    = blockIdx.y;
    const int tid  = threadIdx.x;
    const int lane = tid & 31;
    const int wave = tid >> 5;

    // pre-scale q by 1/sqrt(128) so WMMA emits scaled scores directly
    if (tid < DQ) qs[tid] = q[b * DQ + tid] * 0.08838834764831845f;
    __syncthreads();

    // fp32 A-matrix 16x4 layout: lanes 0-15 -> (M=lane, K=0,1); lanes 16-31 -> (M=lane-16, K=2,3)
    const int   mrow  = lane & 15;
    const int   koff  = (lane >> 4) * 2;
    const size_t krow0 = (size_t)b * SEQLEN;

    float run_max = -3.4028235e38f;
    const int chunk_base = c * CHUNK;

    for (int t = 0; t < TILES_PER_WAVE; ++t) {
        const int s0 = chunk_base + wave * (CHUNK / WAVES) + t * 16;
        const float* arow = k + (krow0 + (size_t)(s0 + mrow)) * DQ + koff;

        v8f acc = {};
        #pragma unroll
        for (int d0 = 0; d0 < DQ; d0 += 4) {
            v2f a  = *(const v2f*)(arow + d0);          // K tile fragment (b64)
            v2f bq = *(const v2f*)(&qs[d0 + koff]);     // q broadcast fragment (ds b64)
            acc = __builtin_amdgcn_wmma_f32_16x16x4_f32(
                false, a, false, bq, (short)0, acc, false, false);
        }

        // column n=0 lives in lanes 0 (M=0..7) and 16 (M=8..15)
        if ((lane & 15) == 0) {
            const int mbase = s0 + ((lane >> 4) << 3);
            #pragma unroll
            for (int j = 0; j < 8; ++j)
                scores_out[(size_t)b * SEQLEN + mbase + j] = acc[j];
        }
        #pragma unroll
        for (int j = 0; j < 8; ++j) run_max = fmaxf(run_max, acc[j]);
    }

    // wave32 max reduction
    #pragma unroll
    for (int off = 16; off > 0; off >>= 1)
        run_max = fmaxf(run_max, __shfl_xor(run_max, off, 32));
    if (lane == 0) wmax[wave] = run_max;
    __syncthreads();
    if (tid == 0) {
        float m = wmax[0];
        #pragma unroll
        for (int i = 1; i < WAVES; ++i) m = fmaxf(m, wmax[i]);
        chunk_max[b * NCHUNK + c] = m;
    }
}

// ---------------------------------------------------------------------------
// Kernel 2: per-batch global max + 1/sum(exp). grid = BATCH, block = 256.
// Scores are L2-resident (2 MB total), so this pass is ~free.
// ---------------------------------------------------------------------------
__global__ __launch_bounds__(256) void
sda_stats_kernel(const float* __restrict__ scores, const float* __restrict__ chunk_max,
                 float* __restrict__ stats)
{
    __shared__ float red[256];
    const int b = blockIdx.x, tid = threadIdx.x;

    float m = -3.4028235e38f;
    if (tid < NCHUNK) m = chunk_max[b * NCHUNK + tid];
    red[tid] = m;
    __syncthreads();
    for (int s = 128; s > 0; s >>= 1) {
        if (tid < s) red[tid] = fmaxf(red[tid], red[tid + s]);
        __syncthreads();
    }
    m = red[0];
    __syncthreads();

    const float* srow = scores + (size_t)b * SEQLEN;
    float acc = 0.0f;
    for (int i = tid; i < SEQLEN; i += 256) acc += expf(srow[i] - m);
    red[tid] = acc;
    __syncthreads();
    for (int s = 128; s > 0; s >>= 1) {
        if (tid < s) red[tid] += red[tid + s];
        __syncthreads();
    }
    if (tid == 0) {
        stats[b * 2 + 0] = m;
        stats[b * 2 + 1] = 1.0f / red[0];
    }
}

// ---------------------------------------------------------------------------
// Kernel 3: final weights + partial attn_vec per chunk.
// grid = (BATCH, NCHUNK), block = 256. V streamed with float4 (full 512B rows).
// ---------------------------------------------------------------------------
__global__ __launch_bounds__(256) void
sda_weights_v_kernel(const float* __restrict__ v, const float* __restrict__ stats,
                     float* __restrict__ attn_w, float* __restrict__ partial)
{
    __shared__ float wlds[CHUNK];
    __shared__ float red[WAVES][DV];

    const int b = blockIdx.x, c = blockIdx.y, tid = threadIdx.x;
    const int lane = tid & 31, wave = tid >> 5;

    const float m    = stats[b * 2 + 0];
    const float invl = stats[b * 2 + 1];

    const size_t sbase = (size_t)b * SEQLEN + (size_t)c * CHUNK;
    for (int i = tid; i < CHUNK; i += 256) {
        const float w = expf(attn_w[sbase + i] - m) * invl;
        attn_w[sbase + i] = w;     // raw score -> final softmax weight
        wlds[i] = w;
    }
    __syncthreads();

    // each wave covers all 128 d-columns (32 lanes x float4), rows s = wave + 8*i
    const int d4 = lane * 4;
    const float* vbase = v + sbase * DV;
    float4 acc = make_float4(0.f, 0.f, 0.f, 0.f);
    for (int s = wave; s < CHUNK; s += WAVES) {
        const float4 vv = *(const float4*)(vbase + (size_t)s * DV + d4);
        const float  w  = wlds[s];
        acc.x += w * vv.x; acc.y += w * vv.y; acc.z += w * vv.z; acc.w += w * vv.w;
    }
    red[wave][d4 + 0] = acc.x; red[wave][d4 + 1] = acc.y;
    red[wave][d4 + 2] = acc.z; red[wave][d4 + 3] = acc.w;
    __syncthreads();

    if (tid < DV) {
        float s = 0.0f;
        #pragma unroll
        for (int w = 0; w < WAVES; ++w) s += red[w][tid];
        partial[((size_t)b * NCHUNK + c) * DV + tid] = s;
    }
}

// ---------------------------------------------------------------------------
// Kernel 4: sum chunk partials -> attn_vec. grid = BATCH, block = 128.
// ---------------------------------------------------------------------------
__global__ __launch_bounds__(128) void
sda_reduce_kernel(const float* __restrict__ partial, float* __restrict__ attn_vec)
{
    const int b = blockIdx.x, d = threadIdx.x;
    float s = 0.0f;
    #pragma unroll
    for (int c = 0; c < NCHUNK; ++c)
        s += partial[((size_t)b * NCHUNK + c) * DV + d];
    attn_vec[b * DV + d] = s;
}

// ---------------------------------------------------------------------------
extern "C" void kernel_launch(void* const* d_in, const int* in_sizes, int n_in,
                              void* d_out, int out_size, void* d_ws, size_t ws_size,
                              hipStream_t stream)
{
    (void)in_sizes; (void)n_in; (void)out_size; (void)ws_size;
    const float* q = (const float*)d_in[0];
    const float* k = (const float*)d_in[1];
    const float* v = (const float*)d_in[2];

    float* out      = (float*)d_out;
    float* attn_vec = out;                    // BATCH*DV floats
    float* attn_w   = out + BATCH * DV;       // BATCH*SEQLEN floats (scores, then weights)

    float* ws        = (float*)d_ws;
    float* chunk_max = ws;                             // BATCH*NCHUNK
    float* stats     = ws + BATCH * NCHUNK;            // 2*BATCH (m, 1/l)
    float* partial   = stats + 2 * BATCH;              // BATCH*NCHUNK*DV

    dim3 gridBC(BATCH, NCHUNK);
    sda_scores_kernel  <<<gridBC, 256, 0, stream>>>(q, k, attn_w, chunk_max);
    sda_stats_kernel   <<<BATCH,  256, 0, stream>>>(attn_w, chunk_max, stats);
    sda_weights_v_kernel<<<gridBC, 256, 0, stream>>>(v, stats, attn_w, partial);
    sda_reduce_kernel  <<<BATCH,  128, 0, stream>>>(partial, attn_vec);
}